// GraphSAGEModel_39986145525988
// MI455X (gfx1250) — compile-verified
//
#include <hip/hip_runtime.h>
#include <hip/hip_bf16.h>

// GraphSAGE (2-layer, mean aggregator) + mean_nodes pool + linear classifier
// for MI455X / gfx1250. f32 WMMA (V_WMMA_F32_16X16X4_F32) used for all GEMMs:
// the workload is scatter/gather memory-bound (~3.5 GB traffic vs 26 GFLOP),
// so native-f32 matrix cores give reference-exact numerics for free.

typedef __attribute__((ext_vector_type(2))) float v2f;
typedef __attribute__((ext_vector_type(8))) float v8f;

#define D_FEAT 128
#define NCLS 16
#define NGRAPH 64

// ---------------------------------------------------------------- utilities
__global__ void k_zero(float* __restrict__ p, int n) {
    int i = blockIdx.x * blockDim.x + threadIdx.x;
    int stride = gridDim.x * blockDim.x;
    for (; i < n; i += stride) p[i] = 0.0f;
}

__global__ void k_degree(const int* __restrict__ dst, float* __restrict__ deg, int E) {
    int i = blockIdx.x * blockDim.x + threadIdx.x;
    if (i < E) atomicAdd(&deg[dst[i]], 1.0f);
}

__global__ void k_gcnt(const int* __restrict__ gid, float* __restrict__ gcnt, int N) {
    int i = blockIdx.x * blockDim.x + threadIdx.x;
    if (i < N) atomicAdd(&gcnt[gid[i]], 1.0f);
}

__global__ void k_invdeg(float* __restrict__ deg, int N) {
    int i = blockIdx.x * blockDim.x + threadIdx.x;
    if (i < N) deg[i] = 1.0f / fmaxf(deg[i], 1.0f);
}

// One wave32 per edge: lane l moves feature elements [4l, 4l+3] (float4 load,
// 4 global_atomic_add_f32). 8 edges per 256-thread block.
__global__ __launch_bounds__(256)
void k_scatter(const float* __restrict__ x, const int* __restrict__ src,
               const int* __restrict__ dst, float* __restrict__ agg, int E) {
    int wave = (blockIdx.x * blockDim.x + threadIdx.x) >> 5;
    int lane = threadIdx.x & 31;
    if (wave >= E) return;
    int s = src[wave];
    int d = dst[wave];
    const float4 v = *(const float4*)(x + (size_t)s * D_FEAT + lane * 4);
    float* o = agg + (size_t)d * D_FEAT + lane * 4;
    atomicAdd(o + 0, v.x);
    atomicAdd(o + 1, v.y);
    atomicAdd(o + 2, v.z);
    atomicAdd(o + 3, v.w);
}

// -------------------------------------------------------- WMMA SAGE layer
// Y[m,n] = act( X @ Wself + (AGG*invdeg) @ Wneigh + bias )
// One wave per 16x16 output tile; block = 256 threads = 8 waves = 8 N-tiles
// of one M-tile row (A operands shared through L1). K=128 in steps of 4 via
// V_WMMA_F32_16X16X4_F32, two independent accumulator chains for ILP.
//
// 32-bit A 16x4 layout (ISA 7.12.2): lanes0-15 hold K={k,k+1} in regs {0,1},
// lanes16-31 hold K={k+2,k+3}. -> koff = (lane>>4)*2. B mirrored (rows of W).
// C/D: reg v, lanes0-15 -> row m0+v, lanes16-31 -> row m0+8+v; col n0+(lane&15).
__global__ __launch_bounds__(256)
void k_sage_gemm(const float* __restrict__ X, const float* __restrict__ AGG,
                 const float* __restrict__ invdeg,
                 const float* __restrict__ Wself, const float* __restrict__ Wneigh,
                 const float* __restrict__ bias,
                 float* __restrict__ Y,          // layer output (or nullptr)
                 float* __restrict__ gsum,       // graph-pool accumulator (or nullptr)
                 const int* __restrict__ gid,
                 int do_relu) {
    const int lane    = threadIdx.x & 31;
    const int l15     = lane & 15;
    const int halfsel = lane >> 4;           // 0 | 1
    const int koff    = halfsel * 2;         // 0 | 2
    const int m0      = blockIdx.x * 16;
    const int n0      = (threadIdx.x >> 5) * 16;
    const int row     = m0 + l15;

    const float idg = invdeg[row];
    const float* xrow = X   + (size_t)row * D_FEAT;
    const float* arow = AGG + (size_t)row * D_FEAT;

    v8f acc_s, acc_n;
    const float bval = bias[n0 + l15];
#pragma unroll
    for (int i = 0; i < 8; ++i) { acc_s[i] = bval; acc_n[i] = 0.0f; }

#pragma unroll 8
    for (int k = 0; k < D_FEAT; k += 4) {
        v2f a, b;
        // self path
        const float2 xs = *(const float2*)(xrow + k + koff);
        a[0] = xs.x; a[1] = xs.y;
        b[0] = Wself[(size_t)(k + koff)     * D_FEAT + n0 + l15];
        b[1] = Wself[(size_t)(k + koff + 1) * D_FEAT + n0 + l15];
        acc_s = __builtin_amdgcn_wmma_f32_16x16x4_f32(false, a, false, b,
                                                      (short)0, acc_s, false, false);
        // neighbor-mean path
        const float2 ns = *(const float2*)(arow + k + koff);
        a[0] = ns.x * idg; a[1] = ns.y * idg;
        b[0] = Wneigh[(size_t)(k + koff)     * D_FEAT + n0 + l15];
        b[1] = Wneigh[(size_t)(k + koff + 1) * D_FEAT + n0 + l15];
        acc_n = __builtin_amdgcn_wmma_f32_16x16x4_f32(false, a, false, b,
                                                      (short)0, acc_n, false, false);
    }

#pragma unroll
    for (int v = 0; v < 8; ++v) {
        const int r = m0 + v + halfsel * 8;
        float val = acc_s[v] + acc_n[v];
        if (do_relu) val = fmaxf(val, 0.0f);
        if (Y)    Y[(size_t)r * D_FEAT + n0 + l15] = val;
        if (gsum) atomicAdd(&gsum[gid[r] * D_FEAT + n0 + l15], val);
    }
}

// ------------------------------------------------- classifier (64x128 @ 128x16)
// 1 block, 4 waves; wave w owns graphs [16w, 16w+16). hg = gsum/cnt on load.
__global__ __launch_bounds__(128)
void k_classifier(const float* __restrict__ gsum, const float* __restrict__ gcnt,
                  const float* __restrict__ Wc, const float* __restrict__ bc,
                  float* __restrict__ out) {
    const int lane    = threadIdx.x & 31;
    const int l15     = lane & 15;
    const int halfsel = lane >> 4;
    const int koff    = halfsel * 2;
    const int m0      = (threadIdx.x >> 5) * 16;
    const int row     = m0 + l15;

    const float inv = 1.0f / fmaxf(gcnt[row], 1.0f);
    const float* grow = gsum + (size_t)row * D_FEAT;

    v8f acc;
    const float bval = bc[l15];
#pragma unroll
    for (int i = 0; i < 8; ++i) acc[i] = bval;

#pragma unroll 8
    for (int k = 0; k < D_FEAT; k += 4) {
        v2f a, b;
        const float2 gs = *(const float2*)(grow + k + koff);
        a[0] = gs.x * inv; a[1] = gs.y * inv;
        b[0] = Wc[(size_t)(k + koff)     * NCLS + l15];
        b[1] = Wc[(size_t)(k + koff + 1) * NCLS + l15];
        acc = __builtin_amdgcn_wmma_f32_16x16x4_f32(false, a, false, b,
                                                    (short)0, acc, false, false);
    }

#pragma unroll
    for (int v = 0; v < 8; ++v)
        out[(size_t)(m0 + v + halfsel * 8) * NCLS + l15] = acc[v];
}

// ---------------------------------------------------------------- launcher
extern "C" void kernel_launch(void* const* d_in, const int* in_sizes, int n_in,
                              void* d_out, int out_size, void* d_ws, size_t ws_size,
                              hipStream_t stream) {
    const float* h   = (const float*)d_in[0];
    const int*   src = (const int*)  d_in[1];
    const int*   dst = (const int*)  d_in[2];
    const int*   gid = (const int*)  d_in[3];
    const float* W1s = (const float*)d_in[5];
    const float* W1n = (const float*)d_in[6];
    const float* b1  = (const float*)d_in[7];
    const float* W2s = (const float*)d_in[8];
    const float* W2n = (const float*)d_in[9];
    const float* b2  = (const float*)d_in[10];
    const float* Wc  = (const float*)d_in[11];
    const float* bc  = (const float*)d_in[12];
    float* out = (float*)d_out;

    const int N = in_sizes[3];   // 100000 (multiple of 16)
    const int E = in_sizes[1];   // 1600000

    // workspace layout
    float* deg  = (float*)d_ws;                        // N
    float* agg  = deg  + N;                            // N*128  (reused layer2)
    float* out1 = agg  + (size_t)N * D_FEAT;           // N*128
    float* gsum = out1 + (size_t)N * D_FEAT;           // 64*128
    float* gcnt = gsum + NGRAPH * D_FEAT;              // 64

    const int nblk256 = 1024;

    // init
    k_zero<<<nblk256, 256, 0, stream>>>(deg, N + N * D_FEAT);     // deg+agg contiguous
    k_zero<<<8, 256, 0, stream>>>(gsum, NGRAPH * D_FEAT + NGRAPH);
    // degree / graph counts
    k_degree<<<(E + 255) / 256, 256, 0, stream>>>(dst, deg, E);
    k_gcnt  <<<(N + 255) / 256, 256, 0, stream>>>(gid, gcnt, N);
    // layer 1: scatter-add of h over edges
    k_scatter<<<(E + 7) / 8, 256, 0, stream>>>(h, src, dst, agg, E);
    k_invdeg<<<(N + 255) / 256, 256, 0, stream>>>(deg, N);
    // layer 1 GEMMs + ReLU -> out1
    k_sage_gemm<<<N / 16, 256, 0, stream>>>(h, agg, deg, W1s, W1n, b1,
                                            out1, nullptr, gid, 1);
    // layer 2: re-zero agg, scatter out1
    k_zero<<<nblk256, 256, 0, stream>>>(agg, N * D_FEAT);
    k_scatter<<<(E + 7) / 8, 256, 0, stream>>>(out1, src, dst, agg, E);
    // layer 2 GEMMs, pooling folded into epilogue (x2 never materialized)
    k_sage_gemm<<<N / 16, 256, 0, stream>>>(out1, agg, deg, W2s, W2n, b2,
                                            nullptr, gsum, gid, 0);
    // mean pool + classifier
    k_classifier<<<1, 128, 0, stream>>>(gsum, gcnt, Wc, bc, out);
}